// AsapEnergy_46059229282940
// MI455X (gfx1250) — compile-verified
//
#include <hip/hip_runtime.h>
#include <hip/hip_bf16.h>

// Problem constants (from reference): B,N,K,L,H = 32, 8192, 16, 256, 1024
#define B_   32
#define N_   8192
#define K_   16
#define L_   256
#define H_   1024
#define N3_  24576      // N*3
#define SP2_ 4          // split-K for fwd GEMM2 (K chunk = 256)
#define SP3_ 48         // split-K for bwd GEMM3 (K chunk = 512)
#define NBLK_E_ (N_ / 256)   // energy blocks per batch = 32

typedef __attribute__((ext_vector_type(2))) float v2f;
typedef __attribute__((ext_vector_type(8))) float v8f;

// D = A(16x4) * B(4x16) + C(16x16), full fp32 via V_WMMA_F32_16X16X4_F32.
__device__ __forceinline__ v8f wmma4(v2f a, v2f b, v8f c) {
  return __builtin_amdgcn_wmma_f32_16x16x4_f32(
      /*neg_a=*/false, a, /*neg_b=*/false, b,
      /*c_mod=*/(short)0, c, /*reuse_a=*/false, /*reuse_b=*/false);
}

// ---------------- reverse-adjacency build (int atomics only) ----------------

__global__ __launch_bounds__(256) void k_zero_cnt(int* __restrict__ cnt) {
  int i = blockIdx.x * 256 + threadIdx.x;
  if (i < N_) cnt[i] = 0;
}

__global__ __launch_bounds__(256) void k_count(
    const int* __restrict__ nbrs, const int* __restrict__ nnb,
    int* __restrict__ cnt) {
  const int n = blockIdx.x * 256 + threadIdx.x;     // n < N_
  const int nn = nnb[n];
  for (int k = 0; k < K_; ++k) {
    if (k >= nn) break;
    atomicAdd(&cnt[nbrs[n * K_ + k]], 1);
  }
}

// Exclusive scan of cnt[8192] -> off[8193]; also seeds cursor[] for fill.
__global__ __launch_bounds__(256) void k_scan(
    const int* __restrict__ cnt, int* __restrict__ off, int* __restrict__ cur) {
  __shared__ int part[256];
  const int t = threadIdx.x;
  const int base = t * 32;
  int loc[32];
  int s = 0;
  for (int i = 0; i < 32; ++i) { loc[i] = s; s += cnt[base + i]; }
  part[t] = s;
  __syncthreads();
  for (int d = 1; d < 256; d <<= 1) {
    int v = (t >= d) ? part[t - d] : 0;
    __syncthreads();
    part[t] += v;
    __syncthreads();
  }
  const int cb = (t == 0) ? 0 : part[t - 1];
  for (int i = 0; i < 32; ++i) {
    off[base + i] = cb + loc[i];
    cur[base + i] = cb + loc[i];
  }
  if (t == 255) off[N_] = part[255];
}

__global__ __launch_bounds__(256) void k_fill(
    const int* __restrict__ nbrs, const int* __restrict__ nnb,
    int* __restrict__ cur, int* __restrict__ rev) {
  const int n = blockIdx.x * 256 + threadIdx.x;
  const int nn = nnb[n];
  for (int k = 0; k < K_; ++k) {
    if (k >= nn) break;
    const int m = nbrs[n * K_ + k];
    const int p = atomicAdd(&cur[m], 1);
    rev[p] = (n << 4) | k;                 // pack (source n, slot k)
  }
}

// ---------------- forward MLP ----------------

// GEMM1: h = relu(code(32x256) @ W1(256x1024) + b1). One wave per 16-col tile.
__global__ __launch_bounds__(32) void k_gemm1_relu(
    const float* __restrict__ code, const float* __restrict__ W1,
    const float* __restrict__ b1, float* __restrict__ h) {
  const int n0   = blockIdx.x * 16;
  const int lane = threadIdx.x;
  const int mrow = lane & 15;
  const int kh   = (lane >> 4) * 2;      // K sub-offset 0 or 2 per lane half
  v8f c0 = {}, c1 = {};
  for (int k = 0; k < L_; k += 4) {
    v2f a0, a1, bb;
    const float* pa0 = code + mrow * L_ + k + kh;
    const float* pa1 = code + (mrow + 16) * L_ + k + kh;
    a0.x = pa0[0]; a0.y = pa0[1];
    a1.x = pa1[0]; a1.y = pa1[1];
    bb.x = W1[(k + kh) * H_ + n0 + mrow];
    bb.y = W1[(k + kh + 1) * H_ + n0 + mrow];
    c0 = wmma4(a0, bb, c0);              // batch rows 0..15
    c1 = wmma4(a1, bb, c1);              // batch rows 16..31
  }
  const int ncol = n0 + mrow;
  const int mb   = (lane >> 4) * 8;      // C/D: lanes>=16 hold M = r+8
  const float bias = b1[ncol];
  for (int r = 0; r < 8; ++r) {
    float v0 = c0[r] + bias;
    float v1 = c1[r] + bias;
    h[(mb + r) * H_ + ncol]      = v0 > 0.0f ? v0 : 0.0f;
    h[(16 + mb + r) * H_ + ncol] = v1 > 0.0f ? v1 : 0.0f;
  }
}

// GEMM2 (split-K x4, 2-wide j tiles): part2[s] = h(32x1024) @ W2 chunk.
// 3072 waves in flight -> ~24 MB outstanding loads, enough to cover
// 23.3 TB/s x ~600ns HBM latency. Plain stores into per-split partials.
__global__ __launch_bounds__(32) void k_gemm2(
    const float* __restrict__ h, const float* __restrict__ W2,
    float* __restrict__ part2) {
  const int j0   = blockIdx.x * 32;
  const int sp   = blockIdx.y;
  const int ks   = sp * (H_ / SP2_);
  const int ke   = ks + (H_ / SP2_);
  const int lane = threadIdx.x;
  const int mrow = lane & 15;
  const int kh   = (lane >> 4) * 2;
  v8f c00 = {}, c01 = {}, c10 = {}, c11 = {};
  for (int k = ks; k < ke; k += 4) {
    if (k + 32 < ke)   // keep W2 stream ahead in L2 (global_prefetch_b8)
      __builtin_prefetch(&W2[(size_t)(k + 32 + kh) * N3_ + j0 + mrow], 0, 1);
    v2f a0, a1, b0, b1v;
    const float* pa0 = h + mrow * H_ + k + kh;
    const float* pa1 = h + (mrow + 16) * H_ + k + kh;
    a0.x = pa0[0]; a0.y = pa0[1];
    a1.x = pa1[0]; a1.y = pa1[1];
    const float* pb = W2 + (size_t)(k + kh) * N3_ + j0 + mrow;
    b0.x  = pb[0];        b0.y  = pb[N3_];          // rows k+kh, k+kh+1
    b1v.x = pb[16];       b1v.y = pb[N3_ + 16];     // second j tile
    c00 = wmma4(a0, b0,  c00);
    c01 = wmma4(a1, b0,  c01);
    c10 = wmma4(a0, b1v, c10);
    c11 = wmma4(a1, b1v, c11);
  }
  float* dst = part2 + (size_t)sp * (B_ * N3_);
  const int jA = j0 + mrow, jB = j0 + 16 + mrow;
  const int mb = (lane >> 4) * 8;
  for (int r = 0; r < 8; ++r) {
    dst[(mb + r) * N3_ + jA]      = c00[r];
    dst[(16 + mb + r) * N3_ + jA] = c01[r];
    dst[(mb + r) * N3_ + jB]      = c10[r];
    dst[(16 + mb + r) * N3_ + jB] = c11[r];
  }
}

// recon = sum_s part2[s] + b2 (broadcast over batch rows).
__global__ __launch_bounds__(256) void k_reduce2(
    const float* __restrict__ part2, const float* __restrict__ b2,
    float* __restrict__ recon) {
  int i = blockIdx.x * 256 + threadIdx.x;
  const int total = B_ * N3_;
  for (; i < total; i += gridDim.x * 256) {
    const int col = i % N3_;
    float s = b2[col];
    for (int p = 0; p < SP2_; ++p) s += part2[(size_t)p * total + i];
    recon[i] = s;
  }
}

// ---------------- energy + grad wrt recon (no f32 atomics) ----------------

// Center term + per-block energy partial. Each thread owns (b,n): plain store.
__global__ __launch_bounds__(256) void k_energy_center(
    const float* __restrict__ xyz, const int* __restrict__ nbrs,
    const int* __restrict__ nnb, const float* __restrict__ wmat,
    const float* __restrict__ recon, float* __restrict__ grad_recon,
    float* __restrict__ me_part) {
  __shared__ float red[256];
  const int t = blockIdx.x * 256 + threadIdx.x;     // t = b*N + n
  const int b = t / N_;
  const int n = t - b * N_;
  const float s = 1.0f / (float)(N_ * K_);
  const float* rb = recon + (size_t)b * N3_;
  const float* xb = xyz   + (size_t)b * N3_;
  const float rx = rb[n * 3 + 0], ry = rb[n * 3 + 1], rz = rb[n * 3 + 2];
  const float xx = xb[n * 3 + 0], xy = xb[n * 3 + 1], xz = xb[n * 3 + 2];
  const int nn = nnb[n];
  float es = 0.0f, gx = 0.0f, gy = 0.0f, gz = 0.0f;
  for (int k = 0; k < K_; ++k) {
    if (k >= nn) break;                 // mask = (k < num_neighbors[n])
    const int   j = nbrs[n * K_ + k];
    const float w = wmat[n * K_ + k];
    const float dx = (rx - rb[j * 3 + 0]) - (xx - xb[j * 3 + 0]);
    const float dy = (ry - rb[j * 3 + 1]) - (xy - xb[j * 3 + 1]);
    const float dz = (rz - rb[j * 3 + 2]) - (xz - xb[j * 3 + 2]);
    es += w * (dx * dx + dy * dy + dz * dz);
    const float c = 2.0f * w * s;
    gx += c * dx; gy += c * dy; gz += c * dz;
  }
  float* gb = grad_recon + (size_t)b * N3_;
  gb[n * 3 + 0] = gx;                   // plain stores (exclusive owner)
  gb[n * 3 + 1] = gy;
  gb[n * 3 + 2] = gz;
  red[threadIdx.x] = es;
  __syncthreads();
  for (int off = 128; off > 0; off >>= 1) {
    if (threadIdx.x < off) red[threadIdx.x] += red[threadIdx.x + off];
    __syncthreads();
  }
  if (threadIdx.x == 0) me_part[blockIdx.x] = red[0];
}

// Neighbor (scatter) term re-expressed as gather over reverse CSR:
// grad_recon[b,m] -= sum_{(n,k): nbrs[n,k]==m} 2w/(NK) * d(b,n,k).
__global__ __launch_bounds__(256) void k_grad_neighbor(
    const float* __restrict__ xyz, const float* __restrict__ wmat,
    const float* __restrict__ recon, const int* __restrict__ off,
    const int* __restrict__ rev, float* __restrict__ grad_recon) {
  const int t = blockIdx.x * 256 + threadIdx.x;     // t = b*N + m
  const int b = t / N_;
  const int m = t - b * N_;
  const float s = 1.0f / (float)(N_ * K_);
  const float* rb = recon + (size_t)b * N3_;
  const float* xb = xyz   + (size_t)b * N3_;
  const float rmx = rb[m * 3 + 0], rmy = rb[m * 3 + 1], rmz = rb[m * 3 + 2];
  const float xmx = xb[m * 3 + 0], xmy = xb[m * 3 + 1], xmz = xb[m * 3 + 2];
  float gx = 0.0f, gy = 0.0f, gz = 0.0f;
  const int e0 = off[m], e1 = off[m + 1];
  for (int e = e0; e < e1; ++e) {
    const int pk = rev[e];
    const int n  = pk >> 4;
    const int k  = pk & 15;
    const float w = wmat[n * K_ + k];
    const float c = 2.0f * w * s;
    const float dx = (rb[n * 3 + 0] - rmx) - (xb[n * 3 + 0] - xmx);
    const float dy = (rb[n * 3 + 1] - rmy) - (xb[n * 3 + 1] - xmy);
    const float dz = (rb[n * 3 + 2] - rmz) - (xb[n * 3 + 2] - xmz);
    gx -= c * dx; gy -= c * dy; gz -= c * dz;
  }
  float* gb = grad_recon + (size_t)b * N3_;
  gb[m * 3 + 0] += gx;                  // exclusive owner -> plain RMW
  gb[m * 3 + 1] += gy;
  gb[m * 3 + 2] += gz;
}

// mean_energy: deterministic 2-stage reduction, no atomics on d_out.
__global__ __launch_bounds__(32) void k_me_final(
    const float* __restrict__ me_part, float* __restrict__ out) {
  const int b = threadIdx.x;            // 32 threads
  float s = 0.0f;
  for (int i = 0; i < NBLK_E_; ++i) s += me_part[b * NBLK_E_ + i];
  out[b] = s * (1.0f / (float)(N_ * K_));
}

// ---------------- backward MLP ----------------

// GEMM3 (bwd, transposed, split-K x48): part3[s] = W2 @ grad_recon^T chunk.
// 6144 waves; W2 likely still L2-resident (100 MB < 192 MB L2) after GEMM2.
__global__ __launch_bounds__(32) void k_gemm3(
    const float* __restrict__ W2, const float* __restrict__ grad_recon,
    float* __restrict__ part3) {
  const int l0 = blockIdx.x * 16;             // M tile over H
  const int b0 = blockIdx.y * 16;             // N tile over batch
  const int sp = blockIdx.z;
  const int ks = sp * (N3_ / SP3_);           // 512-col chunk
  const int lane = threadIdx.x;
  const int mrow = lane & 15;
  const int kh   = (lane >> 4) * 2;
  v8f c = {};
  for (int j = ks; j < ks + (N3_ / SP3_); j += 4) {
    v2f a, bb;
    const float* pa = W2 + (size_t)(l0 + mrow) * N3_ + j + kh;          // A[l][j]
    const float* pb = grad_recon + (size_t)(b0 + mrow) * N3_ + j + kh;  // B[j][b]
    a.x = pa[0];  a.y = pa[1];
    bb.x = pb[0]; bb.y = pb[1];
    c = wmma4(a, bb, c);
  }
  float* dst = part3 + (size_t)sp * (B_ * H_);
  const int bcol = b0 + mrow;
  const int mb   = (lane >> 4) * 8;
  for (int r = 0; r < 8; ++r)
    dst[(size_t)bcol * H_ + l0 + mb + r] = c[r];
}

__global__ __launch_bounds__(256) void k_reduce3(
    const float* __restrict__ part3, float* __restrict__ grad_h) {
  int i = blockIdx.x * 256 + threadIdx.x;
  const int total = B_ * H_;
  if (i < total) {
    float s = 0.0f;
    for (int p = 0; p < SP3_; ++p) s += part3[(size_t)p * total + i];
    grad_h[i] = s;
  }
}

// GEMM4 (bwd, transposed): grad_code^T(256x32) = W1(256x1024) @ (grad_h.relu')^T
__global__ __launch_bounds__(32) void k_gemm4(
    const float* __restrict__ W1, const float* __restrict__ grad_h,
    const float* __restrict__ h, float* __restrict__ grad_code) {
  const int p0 = blockIdx.x * 16;             // M tile over L
  const int b0 = blockIdx.y * 16;             // N tile over batch
  const int lane = threadIdx.x;
  const int mrow = lane & 15;
  const int kh   = (lane >> 4) * 2;
  v8f c = {};
  for (int l = 0; l < H_; l += 4) {
    v2f a, bb;
    const float* pa = W1 + (size_t)(p0 + mrow) * H_ + l + kh;   // W1[p][l]
    const int gi = (b0 + mrow) * H_ + l + kh;
    a.x = pa[0]; a.y = pa[1];
    bb.x = (h[gi + 0] > 0.0f) ? grad_h[gi + 0] : 0.0f;          // relu'
    bb.y = (h[gi + 1] > 0.0f) ? grad_h[gi + 1] : 0.0f;
    c = wmma4(a, bb, c);
  }
  const int bcol = b0 + mrow;
  const int mb   = (lane >> 4) * 8;
  for (int r = 0; r < 8; ++r)
    grad_code[(size_t)bcol * L_ + p0 + mb + r] = c[r];
}

extern "C" void kernel_launch(void* const* d_in, const int* in_sizes, int n_in,
                              void* d_out, int out_size, void* d_ws, size_t ws_size,
                              hipStream_t stream) {
  const float* xyz1          = (const float*)d_in[0];
  const int*   neighbors     = (const int*)  d_in[1];
  const int*   num_neighbors = (const int*)  d_in[2];
  const float* wmat          = (const float*)d_in[3];
  const float* code          = (const float*)d_in[4];
  const float* W1            = (const float*)d_in[5];
  const float* b1            = (const float*)d_in[6];
  const float* W2            = (const float*)d_in[7];
  const float* b2            = (const float*)d_in[8];
  float* out = (float*)d_out;   // [0,32): mean_energy ; [32,8224): code_grad

  // Workspace layout (fully rewritten every launch; no cross-call state).
  float* ws         = (float*)d_ws;
  float* h          = ws;                                // 32*1024
  float* recon      = h + B_ * H_;                       // 32*24576
  float* grad_recon = recon + (size_t)B_ * N3_;          // 32*24576
  float* grad_h     = grad_recon + (size_t)B_ * N3_;     // 32*1024
  float* part2      = grad_h + B_ * H_;                  // 4 * 786432
  float* part3      = part2 + (size_t)SP2_ * B_ * N3_;   // 48 * 32768
  float* me_part    = part3 + (size_t)SP3_ * B_ * H_;    // 1024
  int*   cnt        = (int*)(me_part + B_ * NBLK_E_);    // 8192
  int*   off        = cnt + N_;                          // 8193
  int*   cursor     = off + N_ + 1;                      // 8192
  int*   rev        = cursor + N_;                       // 131072

  // Reverse adjacency (per-launch rebuild; int atomics only).
  k_zero_cnt<<<N_ / 256, 256, 0, stream>>>(cnt);
  k_count<<<N_ / 256, 256, 0, stream>>>(neighbors, num_neighbors, cnt);
  k_scan<<<1, 256, 0, stream>>>(cnt, off, cursor);
  k_fill<<<N_ / 256, 256, 0, stream>>>(neighbors, num_neighbors, cursor, rev);

  // Forward MLP.
  k_gemm1_relu<<<H_ / 16, 32, 0, stream>>>(code, W1, b1, h);
  k_gemm2<<<dim3(N3_ / 32, SP2_), 32, 0, stream>>>(h, W2, part2);
  k_reduce2<<<1536, 256, 0, stream>>>(part2, b2, recon);

  // Energy + analytic grad wrt recon.
  k_energy_center<<<(B_ * N_) / 256, 256, 0, stream>>>(
      xyz1, neighbors, num_neighbors, wmat, recon, grad_recon, me_part);
  k_grad_neighbor<<<(B_ * N_) / 256, 256, 0, stream>>>(
      xyz1, wmat, recon, off, rev, grad_recon);
  k_me_final<<<1, 32, 0, stream>>>(me_part, out);

  // Backward MLP.
  k_gemm3<<<dim3(H_ / 16, 2, SP3_), 32, 0, stream>>>(W2, grad_recon, part3);
  k_reduce3<<<(B_ * H_ + 255) / 256, 256, 0, stream>>>(part3, grad_h);
  k_gemm4<<<dim3(L_ / 16, 2), 32, 0, stream>>>(W1, grad_h, h, out + B_);
}